// WeightedRankingLoss_40166534152819
// MI455X (gfx1250) — compile-verified
//
#include <hip/hip_runtime.h>

// CDNA5 / gfx1250, wave32.
typedef __attribute__((ext_vector_type(2))) float v2f;
typedef __attribute__((ext_vector_type(8))) float v8f;

#define D_DIM 64
#define ROWS_PER_BLOCK 8      // 8 waves of 32 lanes, one row per wave
#define WSECT 384             // 6 sections x 64 floats per wave

// Pass 1: per-row pairwise weighted hinge loss via WMMA bilinear tiles.
//   loss_row = sum over ALL (i,j) of w_ij * relu(sign(x_ij) - x_ij),
//   x_ij = (o_i-o_j)(t_i-t_j) = u_i + u_j - o_i t_j - t_i o_j,  u = o*t.
// (Equals the reference's 2*sum_{i<j}: the term is symmetric, 0 on the diagonal.)
//
// One V_WMMA_F32_16X16X4_F32 produces a full 16x16 tile of x:
//   A row_i = [-o_i, -t_i, u_i, 1],  B col_j = [t_j, o_j, 1, u_j].
__global__ void WeightedRankingLoss_40166534152819_pass1(
    const float* __restrict__ outputs,
    const int*   __restrict__ targets,
    float*       __restrict__ partial,
    int B)
{
    // Per-wave LDS sections (64 floats each):
    //   [0]=-o  [64]=-t  [128]=o  [192]=t  [256]=u  [320]=ones
    __shared__ float sm[ROWS_PER_BLOCK * WSECT];
    __shared__ float sm_red[ROWS_PER_BLOCK];

    const int lane = threadIdx.x & 31;
    const int wave = threadIdx.x >> 5;
    const int row  = blockIdx.x * ROWS_PER_BLOCK + wave;

    float acc = 0.0f;
    unsigned long long winmask = 0ull, backmask = 0ull;
    float* wbase = sm + wave * WSECT;

    if (row < B) {  // wave-uniform: EXEC stays all-ones inside (WMMA-safe)
        const float* orow = outputs + (size_t)row * D_DIM;
        const int*   trow = targets + (size_t)row * D_DIM;

        const float o0 = orow[lane];
        const float o1 = orow[lane + 32];
        const int   t0 = trow[lane];
        const int   t1 = trow[lane + 32];
        const float tf0 = (float)t0, tf1 = (float)t1;

        // Per-index predicate masks: bit k == predicate(targets[row][k]).
        const unsigned wlo = __builtin_amdgcn_ballot_w32(t0 == 1);
        const unsigned whi = __builtin_amdgcn_ballot_w32(t1 == 1);
        const unsigned blo = __builtin_amdgcn_ballot_w32(t0 >= 10);
        const unsigned bhi = __builtin_amdgcn_ballot_w32(t1 >= 10);
        winmask  = ((unsigned long long)whi << 32) | (unsigned long long)wlo;
        backmask = ((unsigned long long)bhi << 32) | (unsigned long long)blo;

        wbase[      lane] = -o0;       wbase[      lane + 32] = -o1;
        wbase[ 64 + lane] = -tf0;      wbase[ 64 + lane + 32] = -tf1;
        wbase[128 + lane] =  o0;       wbase[128 + lane + 32] =  o1;
        wbase[192 + lane] =  tf0;      wbase[192 + lane + 32] =  tf1;
        wbase[256 + lane] =  o0 * tf0; wbase[256 + lane + 32] =  o1 * tf1;
        wbase[320 + lane] =  1.0f;     wbase[320 + lane + 32] =  1.0f;
    }
    __syncthreads();

    if (row < B) {
        const bool lo16 = (lane < 16);
        const int  l15  = lane & 15;
        const int  mi   = lo16 ? 0 : 8;   // C/D: lanes16-31 hold M = r + 8

        // Per-lane operand indices (branchless: pure address selects, hoisted).
        // A v0: lo -o[i] | hi u[i]     A v1: lo -t[i] | hi 1
        // B v0: lo  t[j] | hi 1        B v1: lo  o[j] | hi u[j]
        const int addrA0 = (lo16 ?   0 : 256) + l15;
        const int addrA1 = (lo16 ?  64 : 320) + l15;
        const int addrB0 = (lo16 ? 192 : 320) + l15;
        const int addrB1 = (lo16 ? 128 : 256) + l15;

        // Hoist all A-tile operands (reused across the 4 B tiles).
        float a0v[4], a1v[4];
        unsigned eim[4], bim[4];
        #pragma unroll
        for (int ti = 0; ti < 4; ++ti) {
            a0v[ti] = wbase[addrA0 + ti * 16];
            a1v[ti] = wbase[addrA1 + ti * 16];
            const int ishift = ti * 16 + mi;             // first of 8 i-rows in this lane
            eim[ti] = (unsigned)(winmask  >> ishift) & 0xffu;
            bim[ti] = (unsigned)(backmask >> ishift) & 0xffu;
        }

        #pragma unroll
        for (int tj = 0; tj < 4; ++tj) {
            v2f bmat;
            bmat.x = wbase[addrB0 + tj * 16];
            bmat.y = wbase[addrB1 + tj * 16];

            const int jcol = tj * 16 + l15;
            const unsigned wj = (unsigned)(winmask  >> jcol) & 1u;
            const unsigned bj = (unsigned)(backmask >> jcol) & 1u;
            const unsigned ej8 = wj ? 0xffu : 0u;
            const unsigned bj8 = bj ? 0xffu : 0u;

            #pragma unroll
            for (int ti = 0; ti < 4; ++ti) {
                v2f amat;
                amat.x = a0v[ti];
                amat.y = a1v[ti];

                v8f c = {};
                c = __builtin_amdgcn_wmma_f32_16x16x4_f32(
                        /*neg_a=*/false, amat, /*neg_b=*/false, bmat,
                        /*c_mod=*/(short)0, c, /*reuse_a=*/false, /*reuse_b=*/false);

                // Weight bits for the 8 elements this lane holds:
                //   e  bit r: winner(i)|winner(j)  -> w = 5
                //   zn bit r: both back, not e     -> w = 0, else w = 1
                const unsigned e  = eim[ti] | ej8;
                const unsigned zn = (bim[ti] & bj8) & ~e;

                #pragma unroll
                for (int r = 0; r < 8; ++r) {
                    const float x  = c[r];
                    const float w  = ((e >> r) & 1u) ? 5.0f
                                   : (((zn >> r) & 1u) ? 0.0f : 1.0f);
                    // relu(sign(x)-x) == max(0, (x>0 ? 1 : -1) - x)  (0 at x==0)
                    const float sp = (x > 0.0f) ? 1.0f : -1.0f;
                    const float h  = fmaxf(sp - x, 0.0f);
                    acc = fmaf(w, h, acc);
                }
            }
        }
    }

    // Wave32 tree reduction, then cross-wave via LDS.
    #pragma unroll
    for (int off = 16; off > 0; off >>= 1)
        acc += __shfl_xor(acc, off, 32);
    if (lane == 0) sm_red[wave] = acc;
    __syncthreads();
    if (threadIdx.x == 0) {
        float s = 0.0f;
        #pragma unroll
        for (int w = 0; w < ROWS_PER_BLOCK; ++w) s += sm_red[w];
        partial[blockIdx.x] = s;
    }
}

// Pass 2: reduce block partials, scale by 1/denom, write the scalar output.
__global__ void WeightedRankingLoss_40166534152819_pass2(
    const float* __restrict__ partial, int n,
    float* __restrict__ out, float inv_denom)
{
    __shared__ float sm[8];
    float a = 0.0f;
    for (int i = threadIdx.x; i < n; i += 256)
        a += partial[i];
    #pragma unroll
    for (int off = 16; off > 0; off >>= 1)
        a += __shfl_xor(a, off, 32);
    if ((threadIdx.x & 31) == 0) sm[threadIdx.x >> 5] = a;
    __syncthreads();
    if (threadIdx.x == 0) {
        float s = 0.0f;
        #pragma unroll
        for (int w = 0; w < 8; ++w) s += sm[w];
        out[0] = s * inv_denom;
    }
}

extern "C" void kernel_launch(void* const* d_in, const int* in_sizes, int n_in,
                              void* d_out, int out_size, void* d_ws, size_t ws_size,
                              hipStream_t stream)
{
    const float* outputs = (const float*)d_in[0];
    const int*   targets = (const int*)d_in[1];

    const int B = in_sizes[0] / D_DIM;                                  // 8192
    const int blocks = (B + ROWS_PER_BLOCK - 1) / ROWS_PER_BLOCK;       // 1024

    float* partial = (float*)d_ws;                                      // blocks * 4B

    WeightedRankingLoss_40166534152819_pass1<<<blocks, 256, 0, stream>>>(
        outputs, targets, partial, B);

    const float denom = (float)B * (float)D_DIM * (float)(D_DIM - 1) * 0.5f;
    WeightedRankingLoss_40166534152819_pass2<<<1, 256, 0, stream>>>(
        partial, blocks, (float*)d_out, 1.0f / denom);
}